// GlobalCorrelation_39238821216547
// MI455X (gfx1250) — compile-verified
//
#include <hip/hip_runtime.h>
#include <hip/hip_bf16.h>
#include <math.h>

typedef __attribute__((ext_vector_type(2))) float v2f;
typedef __attribute__((ext_vector_type(4))) float v4f;
typedef __attribute__((ext_vector_type(8))) float v8f;

#define CDIM 128
#define WDIM 192
#define HDIM 96
#define B2DIM 8
#define B1DIM 4
#define ROWS_TOTAL (B2DIM * HDIM * WDIM)   /* 147456 */
#define FLOW_ELEMS (B2DIM * HDIM * WDIM * 2)
#define PSTRIDE 36  /* panel row stride in floats: 144B = 16B-aligned rows,
                       36 mod 64 -> 16 distinct banks over 16 rows */

static __device__ __forceinline__ v8f wmma4(v2f a, v2f b, v8f c) {
  // D = A(16x4,f32) * B(4x16,f32) + C(16x16,f32)
  return __builtin_amdgcn_wmma_f32_16x16x4_f32(false, a, false, b, (short)0, c,
                                               false, false);
}

// async global -> LDS copy of 16 bytes per lane (no VGPR round trip).
// lds_off: byte offset within the workgroup LDS allocation (low 32 bits of
// the generic LDS pointer); gptr: 64-bit global address.
static __device__ __forceinline__ void async_ld_b128(unsigned lds_off,
                                                     const void* gptr) {
  asm volatile("global_load_async_to_lds_b128 %0, %1, off"
               :
               : "v"(lds_off), "v"(gptr)
               : "memory");
}
static __device__ __forceinline__ void wait_async0() {
  asm volatile("s_wait_asynccnt 0x0" ::: "memory");
}

// ---------------------------------------------------------------------------
// Kernel 1: fused LayerNorm(w_norm) + projection (wq for b<4, wk for b>=4)
// block = 128 threads (4 waves), each wave owns a 16-row M-tile.
// ---------------------------------------------------------------------------
__global__ __launch_bounds__(128) void knorm_proj(
    const float* __restrict__ x, const float* __restrict__ wn,
    const float* __restrict__ wq, const float* __restrict__ wk,
    float* __restrict__ proj) {
  __shared__ float xn[64][CDIM + 2];  // +2 pad: conflict-free A-frag reads

  const int lane = threadIdx.x & 31;
  const int wv = threadIdx.x >> 5;
  const int rbase = blockIdx.x * 64 + wv * 16;

  const v4f wnv = *(const v4f*)(wn + lane * 4);

  // per-row layernorm for this wave's 16 rows (lane covers 4 channels)
  for (int r = 0; r < 16; ++r) {
    const float* xr = x + (size_t)(rbase + r) * CDIM;
    v4f v = *(const v4f*)(xr + lane * 4);
    float s = v.x + v.y + v.z + v.w;
    float ss = v.x * v.x + v.y * v.y + v.z * v.z + v.w * v.w;
#pragma unroll
    for (int off = 16; off > 0; off >>= 1) {
      s += __shfl_xor(s, off, 32);
      ss += __shfl_xor(ss, off, 32);
    }
    const float mu = s * (1.0f / CDIM);
    const float var = ss * (1.0f / CDIM) - mu * mu;
    const float rs = rsqrtf(var + 1e-5f);
    float* dst = &xn[wv * 16 + r][lane * 4];
    dst[0] = (v.x - mu) * rs * wnv.x;
    dst[1] = (v.y - mu) * rs * wnv.y;
    dst[2] = (v.z - mu) * rs * wnv.z;
    dst[3] = (v.w - mu) * rs * wnv.w;
  }
  __syncthreads();

  // GEMM: out[row][d] = sum_c xn[row][c] * Wt[d][c]
  const int b = rbase / (HDIM * WDIM);  // 64-row blocks never straddle batches
  const float* Wt = (b < B1DIM) ? wq : wk;
  const int m = lane & 15;
  const int koff = (lane >> 4) << 1;  // 0 or 2

  v8f acc[8];
#pragma unroll
  for (int n = 0; n < 8; ++n)
#pragma unroll
    for (int r = 0; r < 8; ++r) acc[n][r] = 0.0f;

  for (int kb = 0; kb < CDIM; kb += 4) {
    v2f a;
    a.x = xn[wv * 16 + m][kb + koff];
    a.y = xn[wv * 16 + m][kb + koff + 1];
#pragma unroll
    for (int n = 0; n < 8; ++n) {
      // B[k][n] = Wt[n][k]; lane holds N = lane%16, K = kb+koff+{0,1}
      const v2f bf =
          *(const v2f*)(Wt + (size_t)(n * 16 + m) * CDIM + kb + koff);
      acc[n] = wmma4(a, bf, acc[n]);
    }
  }

  const int mbase = (lane >> 4) << 3;  // 0 or 8
#pragma unroll
  for (int n = 0; n < 8; ++n)
#pragma unroll
    for (int r = 0; r < 8; ++r)
      proj[(size_t)(rbase + mbase + r) * CDIM + n * 16 + m] = acc[n][r];
}

// ---------------------------------------------------------------------------
// Kernel 2: per (bq,h): corr GEMM (192x192x128), triangular softmax stats,
// vol_clone for both batch halves. block = 512 threads (16 waves).
// K-panels staged via async global->LDS (ASYNCcnt path, no VGPR round trip).
// ---------------------------------------------------------------------------
__global__ __launch_bounds__(512) void kcorr_vol(
    const float* __restrict__ proj, float* __restrict__ corr_ws,
    float* __restrict__ out_vol) {
  __shared__ float lref[WDIM][PSTRIDE];
  __shared__ float ltgt[WDIM][PSTRIDE];
  __shared__ float sm1[WDIM], srd1[WDIM], sm2[WDIM], srd2[WDIM];

  const int bq = blockIdx.x / HDIM;
  const int h = blockIdx.x % HDIM;
  const int lane = threadIdx.x & 31;
  const int wv = threadIdx.x >> 5;

  const float* ref = proj + (size_t)(bq * HDIM + h) * WDIM * CDIM;
  const float* tgt = proj + (size_t)((bq + B1DIM) * HDIM + h) * WDIM * CDIM;
  float* corr = corr_ws + (size_t)(bq * HDIM + h) * (WDIM * WDIM);

  const int m = lane & 15;
  const int koff = (lane >> 4) << 1;

  v8f acc[9];
#pragma unroll
  for (int t = 0; t < 9; ++t)
#pragma unroll
    for (int r = 0; r < 8; ++r) acc[t][r] = 0.0f;

  for (int kb0 = 0; kb0 < CDIM; kb0 += 32) {
    __syncthreads();  // previous panel fully consumed before overwrite
    // stage 192x32 panels of ref and tgt (1536 float4 each) asynchronously
    for (int i = threadIdx.x; i < 1536; i += 512) {
      const int row = i >> 3;
      const int c4 = (i & 7) << 2;
      async_ld_b128((unsigned)(uintptr_t)&lref[row][c4],
                    ref + (size_t)row * CDIM + kb0 + c4);
      async_ld_b128((unsigned)(uintptr_t)&ltgt[row][c4],
                    tgt + (size_t)row * CDIM + kb0 + c4);
    }
    wait_async0();
    __syncthreads();

#pragma unroll
    for (int t = 0; t < 9; ++t) {
      const int id = wv + t * 16;  // 144 tiles, 9 per wave
      const int it = id / 12;
      const int jt = id % 12;
#pragma unroll
      for (int kk = 0; kk < 32; kk += 4) {
        v2f a, bf;
        a.x = lref[it * 16 + m][kk + koff];
        a.y = lref[it * 16 + m][kk + koff + 1];
        bf.x = ltgt[jt * 16 + m][kk + koff];  // B[k][j] = tgt[j][k]
        bf.y = ltgt[jt * 16 + m][kk + koff + 1];
        acc[t] = wmma4(a, bf, acc[t]);
      }
    }
  }

  const float scale = 0.088388347648318447f;  // 128^-0.5
  const int mbase = (lane >> 4) << 3;
#pragma unroll
  for (int t = 0; t < 9; ++t) {
    const int id = wv + t * 16;
    const int it = id / 12;
    const int jt = id % 12;
#pragma unroll
    for (int r = 0; r < 8; ++r)
      corr[(it * 16 + mbase + r) * WDIM + jt * 16 + m] = acc[t][r] * scale;
  }
  __threadfence();
  __syncthreads();

  // softmax stats over d at each w (invalid entries contribute exp(0))
  const int t = threadIdx.x;
  if (t < WDIM) {
    const int w = t;
    const int ninv = (WDIM - 1) - w;
    float mx = (ninv > 0) ? 0.0f : -1.0e30f;
    const float* rowp = corr + w * WDIM;
    for (int j = 0; j <= w; ++j) mx = fmaxf(mx, rowp[j]);
    float den = (float)ninv * __expf(-mx);
    for (int j = 0; j <= w; ++j) den += __expf(rowp[j] - mx);
    sm1[w] = mx;
    srd1[w] = 1.0f / den;
  } else if (t < 2 * WDIM) {
    const int w = t - WDIM;
    const int ninv = w;
    float mx = (ninv > 0) ? 0.0f : -1.0e30f;
    for (int i = w; i < WDIM; ++i) mx = fmaxf(mx, corr[i * WDIM + w]);
    float den = (float)ninv * __expf(-mx);
    for (int i = w; i < WDIM; ++i) den += __expf(corr[i * WDIM + w] - mx);
    sm2[w] = mx;
    srd2[w] = 1.0f / den;
  }
  __syncthreads();

  // vol_clone for both halves; out layout (B2, D=192, H, W)
  for (int idx = threadIdx.x; idx < 2 * WDIM * WDIM; idx += 512) {
    const int half = idx / (WDIM * WDIM);
    const int rem = idx - half * (WDIM * WDIM);
    const int d = rem / WDIM;
    const int w = rem - d * WDIM;
    float val;
    int bout;
    if (half == 0) {
      const int wp = (w > d) ? w : d;  // max(w,d) -> valid lower-tri entry
      val = __expf(corr[wp * WDIM + (wp - d)] - sm1[wp]) * srd1[wp];
      bout = bq;
    } else {
      const int lim = (WDIM - 1) - d;
      const int wp = (w < lim) ? w : lim;  // min(w, 191-d)
      val = __expf(corr[(wp + d) * WDIM + wp] - sm2[wp]) * srd2[wp];
      bout = bq + B1DIM;
    }
    out_vol[(((size_t)bout * WDIM + d) * HDIM + h) * WDIM + w] = val;
  }
}

// ---------------------------------------------------------------------------
// Kernel 3: box-5 blur over d, first-argmax, threshold + centroid -> flow
// ---------------------------------------------------------------------------
__global__ __launch_bounds__(256) void kflow(const float* __restrict__ vol,
                                             float* __restrict__ flow) {
  const int gid = blockIdx.x * blockDim.x + threadIdx.x;
  if (gid >= ROWS_TOTAL) return;
  const int w = gid % WDIM;
  const int h = (gid / WDIM) % HDIM;
  const int b = gid / (WDIM * HDIM);
  const size_t stride = (size_t)HDIM * WDIM;
  const float* base =
      vol + (size_t)b * WDIM * stride + (size_t)h * WDIM + w;

  // blur[dd] = mean(v[dd-2..dd+2]) with zero pad; argmax = first max
  float a0 = 0.f, a1 = 0.f, a2 = 0.f, a3 = 0.f, a4 = 0.f;
  float rsum = 0.f, best = -1.0f;
  int besti = 0;
  for (int p = 0; p < WDIM + 2; ++p) {
    const float vp = (p < WDIM) ? base[(size_t)p * stride] : 0.0f;
    rsum += vp - a0;
    a0 = a1; a1 = a2; a2 = a3; a3 = a4; a4 = vp;
    const int dd = p - 2;
    if (dd >= 0 && rsum > best) {
      best = rsum;
      besti = dd;
    }
  }
  int mi = besti;
  mi = (mi < 2) ? 2 : ((mi > WDIM - 3) ? (WDIM - 3) : mi);

  float wp[5];
  float ssum = 0.0f;
#pragma unroll
  for (int o = 0; o < 5; ++o) {
    float v = base[(size_t)(mi - 2 + o) * stride];
    v = (v < (1.0f / WDIM)) ? 0.0f : v;
    wp[o] = v;
    ssum += v;
  }
  ssum = fmaxf(ssum, 1.17549435e-38f);
  float disp = 0.0f;
#pragma unroll
  for (int o = 0; o < 5; ++o) disp += (wp[o] / ssum) * (float)(mi - 2 + o);

  const float sign = (b < B1DIM) ? -1.0f : 1.0f;
  flow[(size_t)gid * 2 + 0] = sign * disp;
  flow[(size_t)gid * 2 + 1] = 0.0f;
}

// ---------------------------------------------------------------------------
extern "C" void kernel_launch(void* const* d_in, const int* in_sizes, int n_in,
                              void* d_out, int out_size, void* d_ws,
                              size_t ws_size, hipStream_t stream) {
  const float* x = (const float*)d_in[0];
  const float* wn = (const float*)d_in[1];
  const float* wq = (const float*)d_in[2];
  const float* wk = (const float*)d_in[3];

  float* out = (float*)d_out;
  float* flow = out;                  // 294912 floats
  float* vol = out + FLOW_ELEMS;      // 28311552 floats

  float* proj = (float*)d_ws;                      // 18874368 floats (75.5 MB)
  float* corr = proj + (size_t)ROWS_TOTAL * CDIM;  // 14155776 floats (56.6 MB)

  knorm_proj<<<ROWS_TOTAL / 64, 128, 0, stream>>>(x, wn, wq, wk, proj);
  kcorr_vol<<<B1DIM * HDIM, 512, 0, stream>>>(proj, corr, vol);
  kflow<<<(ROWS_TOTAL + 255) / 256, 256, 0, stream>>>(vol, flow);
}